// GraphSAGE_23630910063248
// MI455X (gfx1250) — compile-verified
//
#include <hip/hip_runtime.h>

typedef float v2f __attribute__((ext_vector_type(2)));
typedef float v8f __attribute__((ext_vector_type(8)));

#define SCAN_T 256

// ---------------- zero int buffer ----------------
__global__ void zero_i(int* __restrict__ p, int n) {
  int i = blockIdx.x * blockDim.x + threadIdx.x;
  if (i < n) p[i] = 0;
}

// ---------------- degree count (int atomics) ----------------
__global__ void deg_count(const int* __restrict__ dst, int* __restrict__ deg, int E) {
  int e = blockIdx.x * blockDim.x + threadIdx.x;
  if (e < E) atomicAdd(&deg[dst[e]], 1);
}

// ---------------- two-level exclusive scan ----------------
__global__ void block_sums(const int* __restrict__ deg, int* __restrict__ bsum, int N) {
  __shared__ int sm[SCAN_T];
  int i = blockIdx.x * SCAN_T + threadIdx.x;
  sm[threadIdx.x] = (i < N) ? deg[i] : 0;
  __syncthreads();
  for (int s = SCAN_T / 2; s > 0; s >>= 1) {
    if (threadIdx.x < s) sm[threadIdx.x] += sm[threadIdx.x + s];
    __syncthreads();
  }
  if (threadIdx.x == 0) bsum[blockIdx.x] = sm[0];
}

__global__ void scan_block_sums(int* __restrict__ bsum, int nblk) {
  // nblk <= 1024 (N=100000 -> 391). Serial LDS scan: trivial time, simple, correct.
  __shared__ int sm[1024];
  int t = threadIdx.x;
  if (t < nblk) sm[t] = bsum[t];
  __syncthreads();
  if (t == 0) {
    int run = 0;
    for (int i = 0; i < nblk; ++i) { int v = sm[i]; sm[i] = run; run += v; }
  }
  __syncthreads();
  if (t < nblk) bsum[t] = sm[t];
}

__global__ void scan_final(const int* __restrict__ deg, const int* __restrict__ bsum,
                           int* __restrict__ rowstart, int* __restrict__ cursor,
                           float* __restrict__ dinv, int N) {
  __shared__ int sm[SCAN_T];
  int i = blockIdx.x * SCAN_T + threadIdx.x;
  int v = (i < N) ? deg[i] : 0;
  sm[threadIdx.x] = v;
  __syncthreads();
  for (int s = 1; s < SCAN_T; s <<= 1) {           // Hillis-Steele inclusive scan
    int add = (threadIdx.x >= s) ? sm[threadIdx.x - s] : 0;
    __syncthreads();
    sm[threadIdx.x] += add;
    __syncthreads();
  }
  if (i < N) {
    int excl = sm[threadIdx.x] - v + bsum[blockIdx.x];
    rowstart[i] = excl;
    cursor[i]   = excl;
    dinv[i] = (v > 0) ? (1.0f / (float)v) : 0.0f;
  }
}

// ---------------- CSR fill: 1 int atomic per edge ----------------
__global__ void csr_fill(const int* __restrict__ src, const int* __restrict__ dst,
                         int* __restrict__ cursor, int* __restrict__ csr, int E) {
  int e = blockIdx.x * blockDim.x + threadIdx.x;
  if (e < E) {
    int d = dst[e];
    int pos = atomicAdd(&cursor[d], 1);
    csr[pos] = src[e];
  }
}

// ---------------- gather-mean, 128 feats: one wave per node, float4/lane ----------------
__global__ void gather_mean128(const float* __restrict__ x, const int* __restrict__ csr,
                               const int* __restrict__ rowstart, const int* __restrict__ deg,
                               const float* __restrict__ dinv, float* __restrict__ hn, int N) {
  int lane = threadIdx.x & 31;
  int n = blockIdx.x * (blockDim.x >> 5) + (threadIdx.x >> 5);
  if (n >= N) return;
  int start = rowstart[n];
  int cnt = deg[n];
  float4 acc = make_float4(0.f, 0.f, 0.f, 0.f);
  for (int j0 = 0; j0 < cnt; j0 += 32) {
    // load 32 neighbor indices at once, broadcast with shfl (1 load per 32 edges)
    int sl = (j0 + lane < cnt) ? csr[start + j0 + lane] : 0;
    int m = min(32, cnt - j0);
    for (int jj = 0; jj < m; ++jj) {
      int s = __shfl(sl, jj);
      const float4 v = ((const float4*)(x + (long)s * 128))[lane];
      acc.x += v.x; acc.y += v.y; acc.z += v.z; acc.w += v.w;
    }
  }
  float sc = dinv[n];
  acc.x *= sc; acc.y *= sc; acc.z *= sc; acc.w *= sc;
  ((float4*)(hn + (long)n * 128))[lane] = acc;
}

// ---------------- gather-mean, F feats (128<F<=160), row stride ld; zeroes pad cols ----------------
__global__ void gather_mean_strided(const float* __restrict__ h, const int* __restrict__ csr,
                                    const int* __restrict__ rowstart, const int* __restrict__ deg,
                                    const float* __restrict__ dinv, float* __restrict__ hn,
                                    int N, int F, int ld) {
  int lane = threadIdx.x & 31;
  int n = blockIdx.x * (blockDim.x >> 5) + (threadIdx.x >> 5);
  if (n >= N) return;
  int start = rowstart[n];
  int cnt = deg[n];
  int fs = 128 + lane;              // scalar tail feature handled by this lane
  bool sv = fs < F;
  float4 acc = make_float4(0.f, 0.f, 0.f, 0.f);
  float accs = 0.f;
  for (int j0 = 0; j0 < cnt; j0 += 32) {
    int sl = (j0 + lane < cnt) ? csr[start + j0 + lane] : 0;
    int m = min(32, cnt - j0);
    for (int jj = 0; jj < m; ++jj) {
      int s = __shfl(sl, jj);
      const float* row = h + (long)s * ld;
      const float4 v = ((const float4*)row)[lane];
      acc.x += v.x; acc.y += v.y; acc.z += v.z; acc.w += v.w;
      if (sv) accs += row[fs];
    }
  }
  float sc = dinv[n];
  float* o = hn + (long)n * ld;
  acc.x *= sc; acc.y *= sc; acc.z *= sc; acc.w *= sc;
  ((float4*)o)[lane] = acc;
  if (sv)           o[fs] = accs * sc;
  else if (fs < ld) o[fs] = 0.f;    // zero K-pad columns so WMMA padding stays exact
}

// ---------------- fused SAGE GEMM: out = act(Aself@Ws + Aneigh@Wn + bias) ----------------
// One wave computes one 16x16 tile with V_WMMA_F32_16X16X4_F32 (full fp32, K step 4).
// K padded to a multiple of 4 with zeroed A columns; B row clamp is memory-safety only.
// EXEC is uniform (all 32 lanes) through every WMMA.
__global__ void sage_gemm_wmma(const float* __restrict__ Aself, int lda_s,
                               const float* __restrict__ Aneigh, int lda_n,
                               const float* __restrict__ Wself,
                               const float* __restrict__ Wneigh,
                               const float* __restrict__ bias,
                               float* __restrict__ out, int ldo,
                               int Mtiles, int Ntiles, int Nrows,
                               int K, int Kvalid, int ncols, int padcols, int relu) {
  int lane = threadIdx.x & 31;
  int wid = blockIdx.x * (blockDim.x >> 5) + (threadIdx.x >> 5);
  if (wid >= Mtiles * Ntiles) return;

  int mt = wid / Ntiles;
  int nt = wid - mt * Ntiles;
  int rowbase = mt * 16;
  int colbase = nt * 16;

  int l15 = lane & 15;
  int hi = lane >> 4;                  // 0: K pair {k,k+1}; 1: {k+2,k+3}
  int col = colbase + l15;
  int cc = min(col, ncols - 1);        // clamp for safety; invalid cols never stored as data
  long arow = min(rowbase + l15, Nrows - 1);

  const float* As = Aself  + arow * (long)lda_s + 2 * hi;
  const float* An = Aneigh + arow * (long)lda_n + 2 * hi;

  v8f acc = {0.f, 0.f, 0.f, 0.f, 0.f, 0.f, 0.f, 0.f};

  for (int k = 0; k < K; k += 4) {
    v2f a0 = *(const v2f*)(As + k);    // A frag: 16x4 fp32 tile
    v2f a1 = *(const v2f*)(An + k);
    int kb = k + 2 * hi;               // B frag: 4x16 fp32 tile (W row-major, ld==ncols)
    long kr0 = min(kb,     Kvalid - 1);
    long kr1 = min(kb + 1, Kvalid - 1);
    v2f b0, b1;
    b0.x = Wself [kr0 * ncols + cc];
    b0.y = Wself [kr1 * ncols + cc];
    b1.x = Wneigh[kr0 * ncols + cc];
    b1.y = Wneigh[kr1 * ncols + cc];
    acc = __builtin_amdgcn_wmma_f32_16x16x4_f32(false, a0, false, b0,
                                                (short)0, acc, false, false);
    acc = __builtin_amdgcn_wmma_f32_16x16x4_f32(false, a1, false, b1,
                                                (short)0, acc, false, false);
  }

  if (col < padcols) {                 // data cols get results, pad cols get zeros
    float bv = bias[cc];
    bool valid = col < ncols;
    int rbase = rowbase + 8 * hi;      // C/D layout: M = v + 8*(lane>=16)
#pragma unroll
    for (int v = 0; v < 8; ++v) {
      int r = rbase + v;
      if (r < Nrows) {
        float val = acc[v] + bv;
        if (relu) val = fmaxf(val, 0.0f);
        out[(long)r * ldo + col] = valid ? val : 0.0f;
      }
    }
  }
}

extern "C" void kernel_launch(void* const* d_in, const int* in_sizes, int n_in,
                              void* d_out, int out_size, void* d_ws, size_t ws_size,
                              hipStream_t stream) {
  const float* x   = (const float*)d_in[0];
  const int*   src = (const int*)  d_in[1];
  const int*   dst = (const int*)  d_in[2];
  const float* Ws1 = (const float*)d_in[3];
  const float* Wn1 = (const float*)d_in[4];
  const float* b1  = (const float*)d_in[5];
  const float* Ws2 = (const float*)d_in[6];
  const float* Wn2 = (const float*)d_in[7];
  const float* b2  = (const float*)d_in[8];
  float* out = (float*)d_out;

  const int INF  = 128;
  const int HID  = in_sizes[5];          // 150
  const int OUTF = in_sizes[8];          // 128
  const int N    = in_sizes[0] / INF;    // 100000
  const int E    = in_sizes[1];          // 1600000
  const int LDH  = (HID + 7) & ~7;       // 152: K%4==0, rows 8B-aligned
  const int Npad = (N + 255) & ~255;
  const int nblk = (N + SCAN_T - 1) / SCAN_T;   // 391 (<=1024)

  // workspace layout
  char* w = (char*)d_ws;
  int* deg_i    = (int*)w;   w += sizeof(int) * (size_t)Npad;
  int* rowstart = (int*)w;   w += sizeof(int) * (size_t)Npad;
  int* cursor   = (int*)w;   w += sizeof(int) * (size_t)Npad;
  int* bsum     = (int*)w;   w += sizeof(int) * 1024;
  int* csr      = (int*)w;   w += sizeof(int) * (size_t)((E + 3) & ~3);
  float* dinv   = (float*)w; w += sizeof(float) * (size_t)Npad;
  float* hn1    = (float*)w; w += sizeof(float) * (size_t)N * INF;
  float* h1     = (float*)w; w += sizeof(float) * (size_t)N * LDH;
  float* hn2    = (float*)w;

  const int T = 256;
  // 1) degree histogram (int atomics) -> exclusive scan -> rowstart/cursor/deg_inv
  zero_i<<<dim3((N + T - 1) / T), dim3(T), 0, stream>>>(deg_i, N);
  deg_count<<<dim3((E + T - 1) / T), dim3(T), 0, stream>>>(dst, deg_i, E);
  block_sums<<<dim3(nblk), dim3(SCAN_T), 0, stream>>>(deg_i, bsum, N);
  scan_block_sums<<<dim3(1), dim3(1024), 0, stream>>>(bsum, nblk);
  scan_final<<<dim3(nblk), dim3(SCAN_T), 0, stream>>>(deg_i, bsum, rowstart, cursor, dinv, N);
  // 2) CSR of incoming edges (1 int atomic per edge)
  csr_fill<<<dim3((E + T - 1) / T), dim3(T), 0, stream>>>(src, dst, cursor, csr, E);
  // 3) layer-1 mean aggregation: register gather-reduce, no FP atomics
  gather_mean128<<<dim3((N + 7) / 8), dim3(T), 0, stream>>>(x, csr, rowstart, deg_i, dinv, hn1, N);
  // 4) layer-1 fused GEMM + bias + relu (fp32 WMMA); writes zeros into h1 pad cols
  {
    int Mtiles = (N + 15) / 16, Ntiles = (HID + 15) / 16;
    long waves = (long)Mtiles * Ntiles;
    sage_gemm_wmma<<<dim3((waves + 7) / 8), dim3(T), 0, stream>>>(
        x, INF, hn1, INF, Ws1, Wn1, b1, h1, LDH,
        Mtiles, Ntiles, N, INF, INF, HID, LDH, 1);
  }
  // 5) layer-2 mean aggregation over hidden feats (writes zeros into hn2 pad cols)
  gather_mean_strided<<<dim3((N + 7) / 8), dim3(T), 0, stream>>>(
      h1, csr, rowstart, deg_i, dinv, hn2, N, HID, LDH);
  // 6) layer-2 fused GEMM + bias -> d_out (K padded to LDH with zeroed A columns)
  {
    int Mtiles = (N + 15) / 16, Ntiles = (OUTF + 15) / 16;
    long waves = (long)Mtiles * Ntiles;
    sage_gemm_wmma<<<dim3((waves + 7) / 8), dim3(T), 0, stream>>>(
        h1, LDH, hn2, LDH, Ws2, Wn2, b2, out, OUTF,
        Mtiles, Ntiles, N, LDH, HID, OUTF, OUTF, 0);
  }
}